// CommunityModel_19267223290042
// MI455X (gfx1250) — compile-verified
//
#include <hip/hip_runtime.h>
#include <hip/hip_bf16.h>
#include <math.h>

// ---------------------------------------------------------------------------
// Problem constants (match reference)
// ---------------------------------------------------------------------------
#define NB      8192          // B: events per batch
#define NR      32            // R: negatives per event
#define ND      256           // D: embedding dim
#define NK      5             // K: communities
#define NNODES  100000        // node table rows
#define NROWS   (NB * (2 + NR))   // 278528 total projected rows

#define LDSTR   260           // padded LDS row stride (floats) to avoid bank conflicts

typedef float v2f __attribute__((ext_vector_type(2)));
typedef float v8f __attribute__((ext_vector_type(8)));

__device__ __forceinline__ float softplusf(float x) {
    return fmaxf(x, 0.0f) + log1pf(expf(-fabsf(x)));
}

// Decode row r of the concatenated [src | dst | neg-flat] row space.
__device__ __forceinline__ void decode_row(int row,
                                           const int* __restrict__ src,
                                           const int* __restrict__ dst,
                                           const int* __restrict__ neg,
                                           const float* __restrict__ ts,
                                           int& node, float& tval) {
    if (row < NB)            { node = src[row];          tval = ts[row]; }
    else if (row < 2 * NB)   { node = dst[row - NB];     tval = ts[row - NB]; }
    else { int j = row - 2 * NB; node = neg[j];          tval = ts[j >> 5]; } // R == 32
}

// Per-row second-stage: h(8 dims/lane) -> 5 logits -> softmax (lane 0 writes).
__device__ __forceinline__ void row_head(const float* __restrict__ W2,
                                         const float* __restrict__ b2,
                                         const float h[8], int lane, int d0,
                                         float* __restrict__ outrow) {
    // Lane's 40 W2 weights are contiguous: W2[d0*5 .. d0*5+40), 16B aligned.
    float w[40];
    const float4* wp = (const float4*)&W2[d0 * NK];
    #pragma unroll
    for (int i = 0; i < 10; ++i) {
        const float4 q = wp[i];
        w[4 * i + 0] = q.x; w[4 * i + 1] = q.y; w[4 * i + 2] = q.z; w[4 * i + 3] = q.w;
    }
    float p[NK] = {0.f, 0.f, 0.f, 0.f, 0.f};
    #pragma unroll
    for (int i = 0; i < 8; ++i)
        #pragma unroll
        for (int k = 0; k < NK; ++k) p[k] += h[i] * w[i * NK + k];
    #pragma unroll
    for (int k = 0; k < NK; ++k)
        #pragma unroll
        for (int off = 16; off > 0; off >>= 1)
            p[k] += __shfl_xor(p[k], off, 32);
    if (lane == 0) {
        float l[NK], m = -1e30f;
        #pragma unroll
        for (int k = 0; k < NK; ++k) { l[k] = p[k] + b2[k]; m = fmaxf(m, l[k]); }
        float s = 0.f;
        #pragma unroll
        for (int k = 0; k < NK; ++k) { l[k] = expf(l[k] - m); s += l[k]; }
        const float inv = 1.0f / s;
        #pragma unroll
        for (int k = 0; k < NK; ++k) outrow[k] = l[k] * inv;
    }
}

// ---------------------------------------------------------------------------
// Pack W1 (256x256 row-major) into k-pair-interleaved layout:
//   W1p[p*512 + n*2 + 0] = W1[(2p  )*256 + n]
//   W1p[p*512 + n*2 + 1] = W1[(2p+1)*256 + n]
// so a B fragment (k=2p+{0,1}, n) is one coalesced global_load_b64.
// ---------------------------------------------------------------------------
__global__ __launch_bounds__(256)
void k_pack_w1(const float* __restrict__ W1, float* __restrict__ W1p) {
    const int p = blockIdx.x;       // 0..127 (k pair)
    const int n = threadIdx.x;      // 0..255
    const float a = W1[(2 * p + 0) * ND + n];
    const float b = W1[(2 * p + 1) * ND + n];
    v2f* dp = (v2f*)&W1p[p * (2 * ND) + n * 2];
    v2f vv; vv.x = a; vv.y = b;
    *dp = vv;
}

// ---------------------------------------------------------------------------
// Kernel 1 (packed): Y = state @ W1, 32-row M tiles, fp32 WMMA 16x16x4.
// 4 waves/WG; wave w owns N cols [64w, 64w+64) for BOTH 16-row halves.
// Inner K-step: 2 ds_load_b64 (A) + 4 coalesced global_load_b64 (B) + 8 WMMA.
// ---------------------------------------------------------------------------
__global__ __launch_bounds__(128)
void k_gemm_y32(const float* __restrict__ state,
                const float* __restrict__ W1p,
                float* __restrict__ Y) {
    __shared__ float As[32 * LDSTR];          // 32x256 A tile, padded stride

    const int t  = threadIdx.x;
    const int m0 = blockIdx.x * 32;

    // Cooperative A-tile load: 32 rows x 64 float4 = 2048 float4, 16/thread.
    #pragma unroll
    for (int i = 0; i < 16; ++i) {
        const int idx4 = i * 128 + t;
        const int row  = idx4 >> 6;
        const int c4   = idx4 & 63;
        const float4 v = *(const float4*)&state[(size_t)(m0 + row) * ND + c4 * 4];
        float* dp = &As[row * LDSTR + c4 * 4];
        dp[0] = v.x; dp[1] = v.y; dp[2] = v.z; dp[3] = v.w;
    }
    __syncthreads();

    const int lane = t & 31;
    const int wave = t >> 5;
    const int lo   = lane & 15;
    const int hi   = lane >> 4;
    const int nb   = wave * 64;

    v8f a0c0 = {}, a0c1 = {}, a0c2 = {}, a0c3 = {};
    v8f a1c0 = {}, a1c1 = {}, a1c2 = {}, a1c3 = {};

    for (int kk = 0; kk < ND; kk += 4) {
        const int k0 = kk + 2 * hi;                    // this lane's k pair base
        v2f a0 = *(const v2f*)&As[lo * LDSTR + k0];            // rows m0..m0+15
        v2f a1 = *(const v2f*)&As[(16 + lo) * LDSTR + k0];     // rows m0+16..m0+31

        const float* bp = &W1p[(size_t)(k0 >> 1) * (2 * ND)];  // k-pair row
        v2f b0 = *(const v2f*)&bp[(nb +  0 + lo) * 2];
        v2f b1 = *(const v2f*)&bp[(nb + 16 + lo) * 2];
        v2f b2 = *(const v2f*)&bp[(nb + 32 + lo) * 2];
        v2f b3 = *(const v2f*)&bp[(nb + 48 + lo) * 2];

        a0c0 = __builtin_amdgcn_wmma_f32_16x16x4_f32(false, a0, false, b0, (short)0, a0c0, false, false);
        a0c1 = __builtin_amdgcn_wmma_f32_16x16x4_f32(false, a0, false, b1, (short)0, a0c1, false, false);
        a0c2 = __builtin_amdgcn_wmma_f32_16x16x4_f32(false, a0, false, b2, (short)0, a0c2, false, false);
        a0c3 = __builtin_amdgcn_wmma_f32_16x16x4_f32(false, a0, false, b3, (short)0, a0c3, false, false);
        a1c0 = __builtin_amdgcn_wmma_f32_16x16x4_f32(false, a1, false, b0, (short)0, a1c0, false, false);
        a1c1 = __builtin_amdgcn_wmma_f32_16x16x4_f32(false, a1, false, b1, (short)0, a1c1, false, false);
        a1c2 = __builtin_amdgcn_wmma_f32_16x16x4_f32(false, a1, false, b2, (short)0, a1c2, false, false);
        a1c3 = __builtin_amdgcn_wmma_f32_16x16x4_f32(false, a1, false, b3, (short)0, a1c3, false, false);
    }

    // C/D layout: element (M = v + 8*hi, N = nb + 16*j + lo) in acc_j[v].
    #pragma unroll
    for (int v = 0; v < 8; ++v) {
        const size_t r0 = (size_t)(m0 + v + 8 * hi) * ND;
        Y[r0 + nb +  0 + lo] = a0c0[v];
        Y[r0 + nb + 16 + lo] = a0c1[v];
        Y[r0 + nb + 32 + lo] = a0c2[v];
        Y[r0 + nb + 48 + lo] = a0c3[v];
        const size_t r1 = (size_t)(m0 + 16 + v + 8 * hi) * ND;
        Y[r1 + nb +  0 + lo] = a1c0[v];
        Y[r1 + nb + 16 + lo] = a1c1[v];
        Y[r1 + nb + 32 + lo] = a1c2[v];
        Y[r1 + nb + 48 + lo] = a1c3[v];
    }
}

// ---------------------------------------------------------------------------
// Kernel 1 (unpacked variant, 16-row tiles) — used when ws has room for Y
// but not the packed W1 copy.
// ---------------------------------------------------------------------------
__global__ __launch_bounds__(128)
void k_gemm_y(const float* __restrict__ state,
              const float* __restrict__ W1,
              float* __restrict__ Y) {
    __shared__ float As[16 * LDSTR];

    const int t  = threadIdx.x;
    const int m0 = blockIdx.x * 16;

    #pragma unroll
    for (int i = 0; i < 8; ++i) {
        const int idx4 = i * 128 + t;
        const int row  = idx4 >> 6;
        const int c4   = idx4 & 63;
        const float4 v = *(const float4*)&state[(size_t)(m0 + row) * ND + c4 * 4];
        float* dp = &As[row * LDSTR + c4 * 4];
        dp[0] = v.x; dp[1] = v.y; dp[2] = v.z; dp[3] = v.w;
    }
    __syncthreads();

    const int lane = t & 31;
    const int wave = t >> 5;
    const int lo   = lane & 15;
    const int hi   = lane >> 4;
    const int nb   = wave * 64;

    v8f acc0 = {}, acc1 = {}, acc2 = {}, acc3 = {};

    for (int kk = 0; kk < ND; kk += 4) {
        v2f a = *(const v2f*)&As[lo * LDSTR + kk + 2 * hi];
        const float* wr0 = &W1[(size_t)(kk + 2 * hi) * ND];
        const float* wr1 = wr0 + ND;
        v2f b0; b0.x = wr0[nb +  0 + lo]; b0.y = wr1[nb +  0 + lo];
        v2f b1; b1.x = wr0[nb + 16 + lo]; b1.y = wr1[nb + 16 + lo];
        v2f b2; b2.x = wr0[nb + 32 + lo]; b2.y = wr1[nb + 32 + lo];
        v2f b3; b3.x = wr0[nb + 48 + lo]; b3.y = wr1[nb + 48 + lo];
        acc0 = __builtin_amdgcn_wmma_f32_16x16x4_f32(false, a, false, b0, (short)0, acc0, false, false);
        acc1 = __builtin_amdgcn_wmma_f32_16x16x4_f32(false, a, false, b1, (short)0, acc1, false, false);
        acc2 = __builtin_amdgcn_wmma_f32_16x16x4_f32(false, a, false, b2, (short)0, acc2, false, false);
        acc3 = __builtin_amdgcn_wmma_f32_16x16x4_f32(false, a, false, b3, (short)0, acc3, false, false);
    }

    #pragma unroll
    for (int v = 0; v < 8; ++v) {
        const size_t mrow = (size_t)(m0 + v + 8 * hi) * ND;
        Y[mrow + nb +  0 + lo] = acc0[v];
        Y[mrow + nb + 16 + lo] = acc1[v];
        Y[mrow + nb + 32 + lo] = acc2[v];
        Y[mrow + nb + 48 + lo] = acc3[v];
    }
}

// ---------------------------------------------------------------------------
// Kernel 2: per-row epilogue reading precomputed Y. One wave per row.
// ---------------------------------------------------------------------------
__global__ __launch_bounds__(256)
void k_epilogue(const int* __restrict__ src, const int* __restrict__ dst,
                const int* __restrict__ neg, const float* __restrict__ ts,
                const float* __restrict__ last_t, const float* __restrict__ log_decay,
                const float* __restrict__ Y, const float* __restrict__ b1,
                const float* __restrict__ W2, const float* __restrict__ b2,
                float* __restrict__ out) {
    const int lane = threadIdx.x & 31;
    const int row  = blockIdx.x * 8 + (threadIdx.x >> 5);

    int node; float tval;
    decode_row(row, src, dst, neg, ts, node, tval);

    const float decay = softplusf(log_decay[0]);
    const float dt    = fmaxf(tval - last_t[node], 0.0f);
    const float gate  = expf(-decay * dt);

    const int d0 = lane * 8;
    const size_t yoff = (size_t)node * ND + d0;
    const float4 y0 = *(const float4*)&Y[yoff];
    const float4 y1 = *(const float4*)&Y[yoff + 4];
    const float4 c0 = *(const float4*)&b1[d0];
    const float4 c1 = *(const float4*)&b1[d0 + 4];

    float h[8];
    h[0] = fmaxf(gate * y0.x + c0.x, 0.0f);
    h[1] = fmaxf(gate * y0.y + c0.y, 0.0f);
    h[2] = fmaxf(gate * y0.z + c0.z, 0.0f);
    h[3] = fmaxf(gate * y0.w + c0.w, 0.0f);
    h[4] = fmaxf(gate * y1.x + c1.x, 0.0f);
    h[5] = fmaxf(gate * y1.y + c1.y, 0.0f);
    h[6] = fmaxf(gate * y1.z + c1.z, 0.0f);
    h[7] = fmaxf(gate * y1.w + c1.w, 0.0f);

    row_head(W2, b2, h, lane, d0, &out[(size_t)row * NK]);
}

// ---------------------------------------------------------------------------
// Fallback: fully fused direct kernel (no workspace).
// ---------------------------------------------------------------------------
__global__ __launch_bounds__(128)
void k_direct(const int* __restrict__ src, const int* __restrict__ dst,
              const int* __restrict__ neg, const float* __restrict__ ts,
              const float* __restrict__ state, const float* __restrict__ last_t,
              const float* __restrict__ log_decay,
              const float* __restrict__ W1, const float* __restrict__ b1,
              const float* __restrict__ W2, const float* __restrict__ b2,
              float* __restrict__ out) {
    __shared__ float As[16 * LDSTR];          // reused later as the H tile
    __shared__ int   s_node[16];
    __shared__ float s_gate[16];

    const int t  = threadIdx.x;
    const int r0 = blockIdx.x * 16;

    if (t < 16) {
        int node; float tval;
        decode_row(r0 + t, src, dst, neg, ts, node, tval);
        const float decay = softplusf(log_decay[0]);
        const float dt    = fmaxf(tval - last_t[node], 0.0f);
        s_node[t] = node;
        s_gate[t] = expf(-decay * dt);
    }
    __syncthreads();

    #pragma unroll
    for (int i = 0; i < 8; ++i) {
        const int idx4 = i * 128 + t;
        const int row  = idx4 >> 6;
        const int c4   = idx4 & 63;
        const float g  = s_gate[row];
        const float4 v = *(const float4*)&state[(size_t)s_node[row] * ND + c4 * 4];
        float* dp = &As[row * LDSTR + c4 * 4];
        dp[0] = g * v.x; dp[1] = g * v.y; dp[2] = g * v.z; dp[3] = g * v.w;
    }
    __syncthreads();

    const int lane = t & 31;
    const int wave = t >> 5;
    const int lo   = lane & 15;
    const int hi   = lane >> 4;
    const int nb   = wave * 64;

    v8f acc0 = {}, acc1 = {}, acc2 = {}, acc3 = {};
    for (int kk = 0; kk < ND; kk += 4) {
        v2f a = *(const v2f*)&As[lo * LDSTR + kk + 2 * hi];
        const float* wr0 = &W1[(size_t)(kk + 2 * hi) * ND];
        const float* wr1 = wr0 + ND;
        v2f f0; f0.x = wr0[nb +  0 + lo]; f0.y = wr1[nb +  0 + lo];
        v2f f1; f1.x = wr0[nb + 16 + lo]; f1.y = wr1[nb + 16 + lo];
        v2f f2; f2.x = wr0[nb + 32 + lo]; f2.y = wr1[nb + 32 + lo];
        v2f f3; f3.x = wr0[nb + 48 + lo]; f3.y = wr1[nb + 48 + lo];
        acc0 = __builtin_amdgcn_wmma_f32_16x16x4_f32(false, a, false, f0, (short)0, acc0, false, false);
        acc1 = __builtin_amdgcn_wmma_f32_16x16x4_f32(false, a, false, f1, (short)0, acc1, false, false);
        acc2 = __builtin_amdgcn_wmma_f32_16x16x4_f32(false, a, false, f2, (short)0, acc2, false, false);
        acc3 = __builtin_amdgcn_wmma_f32_16x16x4_f32(false, a, false, f3, (short)0, acc3, false, false);
    }
    __syncthreads();   // done reading As -> reuse as H tile

    #pragma unroll
    for (int v = 0; v < 8; ++v) {
        const int m = v + 8 * hi;
        As[m * LDSTR + nb +  0 + lo] = fmaxf(acc0[v] + b1[nb +  0 + lo], 0.0f);
        As[m * LDSTR + nb + 16 + lo] = fmaxf(acc1[v] + b1[nb + 16 + lo], 0.0f);
        As[m * LDSTR + nb + 32 + lo] = fmaxf(acc2[v] + b1[nb + 32 + lo], 0.0f);
        As[m * LDSTR + nb + 48 + lo] = fmaxf(acc3[v] + b1[nb + 48 + lo], 0.0f);
    }
    __syncthreads();

    for (int rr = 0; rr < 4; ++rr) {
        const int row = wave * 4 + rr;
        const int d0  = lane * 8;
        const float* hp = &As[row * LDSTR + d0];
        float h[8];
        #pragma unroll
        for (int i = 0; i < 8; ++i) h[i] = hp[i];
        row_head(W2, b2, h, lane, d0, &out[(size_t)(r0 + row) * NK]);
    }
}

// ---------------------------------------------------------------------------
// Launch
// ---------------------------------------------------------------------------
extern "C" void kernel_launch(void* const* d_in, const int* in_sizes, int n_in,
                              void* d_out, int out_size, void* d_ws, size_t ws_size,
                              hipStream_t stream) {
    const int*   src       = (const int*)  d_in[0];
    const int*   dst       = (const int*)  d_in[1];
    const int*   neg       = (const int*)  d_in[2];
    const float* ts        = (const float*)d_in[3];
    // d_in[4] = edge_idxs (unused by reference math)
    const float* state     = (const float*)d_in[5];
    const float* last_t    = (const float*)d_in[6];
    const float* log_decay = (const float*)d_in[7];
    const float* W1        = (const float*)d_in[8];
    const float* b1        = (const float*)d_in[9];
    const float* W2        = (const float*)d_in[10];
    const float* b2        = (const float*)d_in[11];
    float*       out       = (float*)d_out;

    const size_t ybytes   = (size_t)NNODES * ND * sizeof(float);   // 102.4 MB
    const size_t w1pbytes = (size_t)ND * ND * sizeof(float);       // 256 KB

    if (ws_size >= ybytes + w1pbytes) {
        float* Y   = (float*)d_ws;
        float* W1p = (float*)((char*)d_ws + ybytes);
        k_pack_w1<<<ND / 2, ND, 0, stream>>>(W1, W1p);
        k_gemm_y32<<<NNODES / 32, 128, 0, stream>>>(state, W1p, Y);
        k_epilogue<<<NROWS / 8, 256, 0, stream>>>(src, dst, neg, ts, last_t,
                                                  log_decay, Y, b1, W2, b2, out);
    } else if (ws_size >= ybytes) {
        float* Y = (float*)d_ws;
        k_gemm_y<<<NNODES / 16, 128, 0, stream>>>(state, W1, Y);
        k_epilogue<<<NROWS / 8, 256, 0, stream>>>(src, dst, neg, ts, last_t,
                                                  log_decay, Y, b1, W2, b2, out);
    } else {
        k_direct<<<NROWS / 16, 128, 0, stream>>>(src, dst, neg, ts, state, last_t,
                                                 log_decay, W1, b1, W2, b2, out);
    }
}